// EigenGIN_74079595921448
// MI455X (gfx1250) — compile-verified
//
#include <hip/hip_runtime.h>

// ---------------------------------------------------------------------------
// EigenGIN on MI455X (gfx1250, wave32).
//
// Roofline: aggregation dominates bytes (~2.4 GB of L2-resident gather +
// atomic traffic; all feature matrices fit in the 192 MB L2). GEMMs total
// only ~4.5 GFLOP, so we keep full fp32 precision and use
// V_WMMA_F32_16X16X4_F32 (fp32 in / fp32 accumulate) instead of bf16, which
// would quantize the 3-layer residual chain.
//
// GEMM mapping: block = 16-row M tile, wave = 16-col N tile. The 16xK A tile
// (with the GINConv "x_i + sum_j x_j" add fused in) is staged once per block
// in LDS (coalesced float2 global loads, padded stride K+2 for bank-conflict-
// free ds_load_b64 A-fragment reads). K/N/flags are template constants so the
// K loop fully unrolls into immediate-offset loads + WMMA chains.
// ---------------------------------------------------------------------------

#define GIN_NODES 50000
#define GIN_EDGES 800000

typedef __attribute__((ext_vector_type(2))) float v2f;
typedef __attribute__((ext_vector_type(8))) float v8f;

// ----------------------------- zero scratch --------------------------------
__global__ void gin_zero_kernel(float* __restrict__ p, int n) {
    int i = blockIdx.x * blockDim.x + threadIdx.x;
    if (i < n) p[i] = 0.0f;
}

// ------------------------- edge scatter-aggregate --------------------------
// One lane per (edge, feature) element: 'dim' consecutive lanes handle one
// edge, so atomics land on consecutive addresses of the destination row
// (coalesced cacheline atomics resolved in L2).
__global__ void gin_edge_agg_kernel(const float* __restrict__ feat,
                                    const int* __restrict__ src,
                                    const int* __restrict__ dst,
                                    float* __restrict__ agg,
                                    int n_edges, int dim_shift) {
    const int mask = (1 << dim_shift) - 1;
    const int total  = n_edges << dim_shift;          // 102.4M max, fits i32
    const int stride = gridDim.x * blockDim.x;
    for (int i = blockIdx.x * blockDim.x + threadIdx.x; i < total; i += stride) {
        const int e = i >> dim_shift;
        const int f = i & mask;
        const int s = src[e];
        const int d = dst[e];
        // global_atomic_add_f32 (no CAS loop)
        unsafeAtomicAdd(&agg[(d << dim_shift) + f], feat[(s << dim_shift) + f]);
    }
}

// ----------------------- fused GIN layer GEMM (WMMA) -----------------------
// Out[M,N] = act( (A + Agg)[M,K] @ W[K,N] + bias ) + Res
//
// ISA VGPR layouts (CDNA5 ISA, V_WMMA_F32_16X16X4_F32):
//   A 16x4 f32 : lane<16 -> M=lane, K={0,1}; lane>=16 -> M=lane-16, K={2,3}
//   B 4x16 f32 : VGPR g, lane<16 -> K=g, N=lane; lane>=16 -> K=g+2, N=lane-16
//   C/D 16x16  : VGPR g, lane<16 -> M=g, N=lane; lane>=16 -> M=g+8, N=lane-16
template <int K, int N, bool AGG, bool RES, bool RELU>
__global__ void __launch_bounds__(N * 2) gin_gemm_wmma_kernel(
        const float* __restrict__ A,     // M x K
        const float* __restrict__ Agg,   // M x K   (used iff AGG)
        const float* __restrict__ W,     // K x N   (row-major)
        const float* __restrict__ bias,  // N
        const float* __restrict__ Res,   // M x N   (used iff RES)
        float* __restrict__ Out) {       // M x N
    constexpr int SK = K + 2;            // padded LDS row stride (8B aligned,
                                         // bank = 2*row mod 64: conflict-free)
    __shared__ float sA[16 * SK];

    const int tid   = threadIdx.x;
    const int lane  = tid & 31;
    const int ntile = tid >> 5;                  // one 16-col tile per wave
    const int r     = lane & 15;                 // A row within tile
    const int kh    = lane >> 4;                 // K-half select (0/1)
    const int row0  = blockIdx.x * 16;
    const int col   = ntile * 16 + (lane & 15);  // global output column

    // ---- stage 16xK A tile (+ fused agg add) into LDS, coalesced float2 ----
    {
        constexpr int K2 = K / 2;                // float2 per row
        const float2* __restrict__ gA = (const float2*)(A + row0 * K);
        const float2* __restrict__ gG = (const float2*)(Agg + row0 * K);
#pragma unroll
        for (int i = tid; i < 8 * K; i += N * 2) {   // 16*K floats as float2
            const int row = i / K2;
            const int k   = (i - row * K2) * 2;
            float2 v = gA[i];
            if (AGG) {
                const float2 g = gG[i];
                v.x += g.x;
                v.y += g.y;
            }
            *(float2*)&sA[row * SK + k] = v;         // ds_store_b64
        }
    }
    __syncthreads();

    // ---- K loop: fully unrolled fp32 WMMA chain ----
    const float* __restrict__ Wc = W + col;          // + compile-time ka*N
    const float* __restrict__ sAr = &sA[r * SK];     // + compile-time ka
    v8f c = {};
#pragma unroll
    for (int k0 = 0; k0 < K; k0 += 4) {
        const int ka = k0 + (kh << 1);
        const v2f a = *(const v2f*)&sAr[ka];         // ds_load_b64
        v2f b;
        b.x = Wc[ka * N];                            // imm-offset global loads
        b.y = Wc[(ka + 1) * N];
        // 8 args: (neg_a, A, neg_b, B, c_mod, C, reuse_a, reuse_b)
        c = __builtin_amdgcn_wmma_f32_16x16x4_f32(
                false, a, false, b, (short)0, c, false, false);
    }

    // ---- epilogue: bias + ReLU + residual, per C/D layout ----
    const float bv = bias[col];
#pragma unroll
    for (int g = 0; g < 8; ++g) {
        const int row = row0 + g + (kh << 3);        // C layout M index
        float v = c[g] + bv;
        if (RELU) v = fmaxf(v, 0.0f);
        const int oidx = row * N + col;              // < 6.4M, i32 math
        if (RES) v += Res[oidx];                     // GIN residual
        Out[oidx] = v;
    }
}

// ---------------------------------------------------------------------------
extern "C" void kernel_launch(void* const* d_in, const int* in_sizes, int n_in,
                              void* d_out, int out_size, void* d_ws, size_t ws_size,
                              hipStream_t stream) {
    (void)in_sizes; (void)n_in; (void)out_size; (void)ws_size;

    const float* x  = (const float*)d_in[0];   // 50000 x 64
    const int*   ei = (const int*)d_in[1];     // 2 x 800000 (int32 per harness)
    const float* W0 = (const float*)d_in[2];   // 64 x 128
    const float* b0 = (const float*)d_in[3];
    const float* W1 = (const float*)d_in[4];   // 128 x 128
    const float* b1 = (const float*)d_in[5];
    const float* W2 = (const float*)d_in[6];   // 128 x 128
    const float* b2 = (const float*)d_in[7];
    const float* Wp = (const float*)d_in[8];   // 128 x 32
    const float* bp = (const float*)d_in[9];
    float* out = (float*)d_out;                // 50000 x 32

    const int* src = ei;                       // edge_index[0]
    const int* dst = ei + GIN_EDGES;           // edge_index[1]

    float* agg = (float*)d_ws;                        // 50000*128 f32
    float* hA  = agg + (size_t)GIN_NODES * 128;       // 50000*128 f32
    float* hB  = hA  + (size_t)GIN_NODES * 128;       // 50000*128 f32

    const int MT = GIN_NODES / 16;             // 3125 exact M tiles
    const dim3 aggGrid(2048), aggBlk(256);

    // ---- layer 0: h = relu((x + agg(x)) @ W0 + b0), K=64 -> hA ----
    {
        const int n = GIN_NODES * 64;
        gin_zero_kernel<<<(n + 255) / 256, 256, 0, stream>>>(agg, n);
        gin_edge_agg_kernel<<<aggGrid, aggBlk, 0, stream>>>(x, src, dst, agg,
                                                            GIN_EDGES, 6);
        gin_gemm_wmma_kernel<64, 128, true, false, true>
            <<<MT, 256, 0, stream>>>(x, agg, W0, b0, nullptr, hA);
    }
    // ---- layer 1: h = relu((h + agg(h)) @ W1 + b1) + h -> hB ----
    {
        const int n = GIN_NODES * 128;
        gin_zero_kernel<<<(n + 255) / 256, 256, 0, stream>>>(agg, n);
        gin_edge_agg_kernel<<<aggGrid, aggBlk, 0, stream>>>(hA, src, dst, agg,
                                                            GIN_EDGES, 7);
        gin_gemm_wmma_kernel<128, 128, true, true, true>
            <<<MT, 256, 0, stream>>>(hA, agg, W1, b1, hA, hB);
    }
    // ---- layer 2: h = relu((h + agg(h)) @ W2 + b2) + h -> hA ----
    {
        const int n = GIN_NODES * 128;
        gin_zero_kernel<<<(n + 255) / 256, 256, 0, stream>>>(agg, n);
        gin_edge_agg_kernel<<<aggGrid, aggBlk, 0, stream>>>(hB, src, dst, agg,
                                                            GIN_EDGES, 7);
        gin_gemm_wmma_kernel<128, 128, true, true, true>
            <<<MT, 256, 0, stream>>>(hB, agg, W2, b2, hB, hA);
    }
    // ---- projection: out = h @ Wp + bp (N=32 -> 2 waves/block) ----
    gin_gemm_wmma_kernel<128, 32, false, false, false>
        <<<MT, 64, 0, stream>>>(hA, nullptr, Wp, bp, nullptr, out);
}